// conv_Update_75419625717909
// MI455X (gfx1250) — compile-verified
//
#include <hip/hip_runtime.h>

typedef float v2f __attribute__((ext_vector_type(2)));
typedef float v4f __attribute__((ext_vector_type(4)));
typedef float v8f __attribute__((ext_vector_type(8)));
typedef int   v4i __attribute__((ext_vector_type(4)));

#define Hh 96
#define Ww 96
#define HW (Hh*Ww)          // 9216
#define Bb 4
#define NPIX (Bb*HW)        // 36864
#define EPSX 1e-5f

__device__ __forceinline__ float sigmoidf_(float v) {
    return 1.0f / (1.0f + __expf(-v));
}

// ---------------------------------------------------------------------------
// Concat [x,msg1,msg2,msg3] along channels -> inp[B][128][HW]
// ---------------------------------------------------------------------------
__global__ __launch_bounds__(256)
void concat_k(const float* __restrict__ x,  const float* __restrict__ m1,
              const float* __restrict__ m2, const float* __restrict__ m3,
              float* __restrict__ inp)
{
    int i = blockIdx.x * 256 + threadIdx.x;
    const int total = Bb * 128 * HW;
    if (i >= total) return;
    int p = i % HW;
    int c = (i / HW) % 128;
    int b = i / (128 * HW);
    const float* s = (c < 32) ? x : (c < 64) ? m1 : (c < 96) ? m2 : m3;
    inp[i] = s[((size_t)b * 32 + (c & 31)) * HW + p];
}

// ---------------------------------------------------------------------------
// Repack weights OIHW (O,C,3,3) -> k-major K order, padded rows:
//   out[o][k*C + c] = (o < O) ? in[(o*C + c)*9 + k] : 0
// ---------------------------------------------------------------------------
__global__ __launch_bounds__(256)
void repack_w(const float* __restrict__ in, float* __restrict__ out,
              int O, int Opad, int C)
{
    int idx = blockIdx.x * 256 + threadIdx.x;
    int KD = C * 9;
    if (idx >= Opad * KD) return;
    int o = idx / KD;
    int r = idx - o * KD;       // r = k*C + c
    int k = r / C;
    int c = r - k * C;
    out[idx] = (o < O) ? in[(o * C + c) * 9 + k] : 0.0f;
}

// ---------------------------------------------------------------------------
// Unified (deformable-)conv GEMM via V_WMMA_F32_16X16X4_F32.
//   out[b][o][p] = sum_{c,k} sample(src, b, c, k, p) * w[o][c][k]  (+bias)
// K dimension ordered k-major (K = k*CIN + c) to match repacked weights:
// metadata is k-only -> hoisted out of the inner channel loop.
// DEFORM=false: plain zero-padded conv3x3 (dy=dx=0, mask=1).
// Grid: (HW/16/8, B); one wave = one 16-pixel N-tile, all OTILES M-tiles.
// ---------------------------------------------------------------------------
template<int CIN, int OTILES, int OSTORE, bool DEFORM>
__global__ __launch_bounds__(256)
void mdconv_gemm(const float* __restrict__ src,
                 const float* __restrict__ om,
                 const float* __restrict__ wgt,   // [OTILES*16][9*CIN] repacked
                 const float* __restrict__ bias,
                 float* __restrict__ dst)
{
    __shared__ __align__(16) int smeta[8][9 * 16 * 8];

    const int lane  = threadIdx.x & 31;
    const int wv    = threadIdx.x >> 5;
    const int b     = blockIdx.y;                       // scalar batch index
    const int nbase = (blockIdx.x * 8 + wv) * 16;       // pixel base in image

    const float* __restrict__ srcb = src + (size_t)b * CIN * HW;   // scalar base

    // ---- per-(k,pixel) sampling metadata: lanes 0..15 fill 9 entries ------
    if (lane < 16) {
        const int p = nbase + lane;
        const int y = p / Ww;
        const int x = p - y * Ww;
        const float* omb = DEFORM ? (om + (size_t)b * 27 * HW + p) : nullptr;
        for (int k = 0; k < 9; ++k) {
            float dy = 0.0f, dx = 0.0f, msk = 1.0f;
            if (DEFORM) {
                dy  = omb[(2 * k    ) * HW];
                dx  = omb[(2 * k + 1) * HW];
                msk = sigmoidf_(omb[(18 + k) * HW]);
            }
            float py = (float)(y + k / 3 - 1) + dy;
            float px = (float)(x + k % 3 - 1) + dx;
            float y0f = floorf(py), x0f = floorf(px);
            float wy1 = py - y0f,  wx1 = px - x0f;
            float wy0 = 1.0f - wy1, wx0 = 1.0f - wx1;
            int y0 = (int)y0f, x0 = (int)x0f;
            int y1 = y0 + 1,   x1 = x0 + 1;
            float vy0 = (y0 >= 0 && y0 < Hh) ? 1.0f : 0.0f;
            float vy1 = (y1 >= 0 && y1 < Hh) ? 1.0f : 0.0f;
            float vx0 = (x0 >= 0 && x0 < Ww) ? 1.0f : 0.0f;
            float vx1 = (x1 >= 0 && x1 < Ww) ? 1.0f : 0.0f;
            int yc0 = min(max(y0, 0), Hh - 1), yc1 = min(max(y1, 0), Hh - 1);
            int xc0 = min(max(x0, 0), Ww - 1), xc1 = min(max(x1, 0), Ww - 1);
            int* e = &smeta[wv][(k * 16 + lane) * 8];
            e[0] = yc0 * Ww + xc0;
            e[1] = yc0 * Ww + xc1;
            e[2] = yc1 * Ww + xc0;
            e[3] = yc1 * Ww + xc1;
            e[4] = __float_as_int(wy0 * wx0 * msk * vy0 * vx0);
            e[5] = __float_as_int(wy0 * wx1 * msk * vy0 * vx1);
            e[6] = __float_as_int(wy1 * wx0 * msk * vy1 * vx0);
            e[7] = __float_as_int(wy1 * wx1 * msk * vy1 * vx1);
        }
    }
    __syncthreads();

    // ---- per-lane fixed coordinates --------------------------------------
    const int ml    = lane & 15;
    const int pl    = nbase + ml;
    const int khalf = (lane >> 4) << 1;   // lanes 0-15 -> K+{0,1}; 16-31 -> K+{2,3}
    const int cO    = khalf * HW;         // plane offset of this lane's first element
    const int* metaw = &smeta[wv][0];
    const int KD = CIN * 9;

    v8f acc[OTILES];
    const v8f zero = {0.f, 0.f, 0.f, 0.f, 0.f, 0.f, 0.f, 0.f};
#pragma unroll
    for (int m = 0; m < OTILES; ++m) acc[m] = zero;

    int arow[OTILES];
#pragma unroll
    for (int m = 0; m < OTILES; ++m) arow[m] = (m * 16 + ml) * KD + khalf;

    // ---- K loop: outer tap k (meta hoisted), inner channel groups of 4 ----
    for (int k = 0; k < 9; ++k) {
        const int* e = metaw + (k * 16 + ml) * 8;
        const v4i o = *(const v4i*)e;
        const v4f w = *(const v4f*)(e + 4);
        const int abase = k * CIN;
        int voff = cO;
#pragma unroll 2
        for (int cs = 0; cs < CIN / 4; ++cs) {
            v2f bf;
            bf.x = w.x * srcb[voff + o.x] + w.y * srcb[voff + o.y]
                 + w.z * srcb[voff + o.z] + w.w * srcb[voff + o.w];
            bf.y = w.x * srcb[voff + HW + o.x] + w.y * srcb[voff + HW + o.y]
                 + w.z * srcb[voff + HW + o.z] + w.w * srcb[voff + HW + o.w];
#pragma unroll
            for (int m = 0; m < OTILES; ++m) {
                const v2f af = *(const v2f*)(wgt + arow[m] + abase + 4 * cs);
                acc[m] = __builtin_amdgcn_wmma_f32_16x16x4_f32(
                    false, af, false, bf, (short)0, acc[m], false, false);
            }
            voff += 4 * HW;
        }
    }

    // ---- epilogue: D layout lane(0-15)=N, vgpr r -> M=r(+8 for hi lanes) --
    const int hi8 = (lane >> 4) * 8;
    float* dstb = dst + (size_t)b * OSTORE * HW;
#pragma unroll
    for (int m = 0; m < OTILES; ++m) {
#pragma unroll
        for (int r = 0; r < 8; ++r) {
            const int Mv = m * 16 + r + hi8;
            if (Mv < OSTORE) {
                float v = acc[m][r];
                if (!DEFORM) v += bias[Mv];
                dstb[Mv * HW + pl] = v;
            }
        }
    }
}

// ---------------------------------------------------------------------------
// BatchNorm (batch stats over B,H,W) + ReLU.  One block per channel.
// ---------------------------------------------------------------------------
template<int CH>
__global__ __launch_bounds__(256)
void bn_relu_k(const float* __restrict__ in, const float* __restrict__ g,
               const float* __restrict__ bt, float* __restrict__ out)
{
    __shared__ float rs[256], rq[256];
    const int c = blockIdx.x;
    float s = 0.f, q = 0.f;
    for (int i = threadIdx.x; i < Bb * HW; i += 256) {
        int b = i / HW; int p = i - b * HW;
        float v = in[((size_t)b * CH + c) * HW + p];
        s += v; q += v * v;
    }
    rs[threadIdx.x] = s; rq[threadIdx.x] = q;
    __syncthreads();
    for (int off = 128; off > 0; off >>= 1) {
        if (threadIdx.x < off) {
            rs[threadIdx.x] += rs[threadIdx.x + off];
            rq[threadIdx.x] += rq[threadIdx.x + off];
        }
        __syncthreads();
    }
    const float inv  = 1.0f / (float)(Bb * HW);
    const float mean = rs[0] * inv;
    const float var  = rq[0] * inv - mean * mean;
    const float sc   = g[c] * rsqrtf(var + EPSX);
    const float sh   = bt[c] - mean * sc;
    for (int i = threadIdx.x; i < Bb * HW; i += 256) {
        int b = i / HW; int p = i - b * HW;
        size_t idx = ((size_t)b * CH + c) * HW + p;
        out[idx] = fmaxf(0.0f, in[idx] * sc + sh);
    }
}

// Final: out = relu(gamma*x + relu(bn(in)))   (CH = 32)
__global__ __launch_bounds__(256)
void bn_final_k(const float* __restrict__ in, const float* __restrict__ g,
                const float* __restrict__ bt, const float* __restrict__ x,
                const float* __restrict__ gamma, float* __restrict__ out)
{
    __shared__ float rs[256], rq[256];
    const int CH = 32;
    const int c = blockIdx.x;
    float s = 0.f, q = 0.f;
    for (int i = threadIdx.x; i < Bb * HW; i += 256) {
        int b = i / HW; int p = i - b * HW;
        float v = in[((size_t)b * CH + c) * HW + p];
        s += v; q += v * v;
    }
    rs[threadIdx.x] = s; rq[threadIdx.x] = q;
    __syncthreads();
    for (int off = 128; off > 0; off >>= 1) {
        if (threadIdx.x < off) {
            rs[threadIdx.x] += rs[threadIdx.x + off];
            rq[threadIdx.x] += rq[threadIdx.x + off];
        }
        __syncthreads();
    }
    const float inv  = 1.0f / (float)(Bb * HW);
    const float mean = rs[0] * inv;
    const float var  = rq[0] * inv - mean * mean;
    const float sc   = g[c] * rsqrtf(var + EPSX);
    const float sh   = bt[c] - mean * sc;
    const float gm   = gamma[0];
    for (int i = threadIdx.x; i < Bb * HW; i += 256) {
        int b = i / HW; int p = i - b * HW;
        size_t idx = ((size_t)b * CH + c) * HW + p;
        float t = fmaxf(0.0f, in[idx] * sc + sh);
        out[idx] = fmaxf(0.0f, gm * x[idx] + t);
    }
}

// ---------------------------------------------------------------------------
extern "C" void kernel_launch(void* const* d_in, const int* in_sizes, int n_in,
                              void* d_out, int out_size, void* d_ws, size_t ws_size,
                              hipStream_t stream)
{
    const float* x     = (const float*)d_in[0];
    const float* m1    = (const float*)d_in[1];
    const float* m2    = (const float*)d_in[2];
    const float* m3    = (const float*)d_in[3];
    const float* offw1 = (const float*)d_in[4];
    const float* offb1 = (const float*)d_in[5];
    const float* w1    = (const float*)d_in[6];
    const float* g1    = (const float*)d_in[7];
    const float* b1    = (const float*)d_in[8];
    const float* offw2 = (const float*)d_in[9];
    const float* offb2 = (const float*)d_in[10];
    const float* w2    = (const float*)d_in[11];
    const float* g2    = (const float*)d_in[12];
    const float* b2    = (const float*)d_in[13];
    const float* gamma = (const float*)d_in[14];

    float* ws = (float*)d_ws;
    size_t o = 0;
    float* inp    = ws + o; o += (size_t)Bb * 128 * HW;   // 18.9 MB
    float* om1    = ws + o; o += (size_t)Bb * 27 * HW;    //  4.0 MB
    float* gA     = ws + o; o += (size_t)Bb * 64 * HW;    //  9.4 MB
    float* h      = ws + o; o += (size_t)Bb * 64 * HW;    //  9.4 MB
    float* om2    = ws + o; o += (size_t)Bb * 27 * HW;    //  4.0 MB
    float* gB     = ws + o; o += (size_t)Bb * 32 * HW;    //  4.7 MB
    float* offw1p = ws + o; o += (size_t)32 * 1152;       // padded+repacked
    float* w1p    = ws + o; o += (size_t)64 * 1152;
    float* offw2p = ws + o; o += (size_t)32 * 576;
    float* w2p    = ws + o; o += (size_t)32 * 576;

    // weight repacks (k-major K order, zero-padded to 32 rows for offset convs)
    repack_w<<<(32 * 1152 + 255) / 256, 256, 0, stream>>>(offw1, offw1p, 27, 32, 128);
    repack_w<<<(64 * 1152 + 255) / 256, 256, 0, stream>>>(w1,    w1p,    64, 64, 128);
    repack_w<<<(32 * 576  + 255) / 256, 256, 0, stream>>>(offw2, offw2p, 27, 32, 64);
    repack_w<<<(32 * 576  + 255) / 256, 256, 0, stream>>>(w2,    w2p,    32, 32, 64);

    const int nconcat = Bb * 128 * HW;
    concat_k<<<(nconcat + 255) / 256, 256, 0, stream>>>(x, m1, m2, m3, inp);

    const dim3 GG(HW / 16 / 8, Bb);   // (72, 4): 8 waves/block, 1 tile/wave

    // stage 1: offset conv (128 -> 27, padded 32), deform GEMM (9*128 -> 64)
    mdconv_gemm<128, 2, 27, false><<<GG, 256, 0, stream>>>(inp, nullptr, offw1p, offb1, om1);
    mdconv_gemm<128, 4, 64, true ><<<GG, 256, 0, stream>>>(inp, om1,     w1p,    nullptr, gA);
    bn_relu_k<64><<<64, 256, 0, stream>>>(gA, g1, b1, h);

    // stage 2: offset conv (64 -> 27, padded 32), deform GEMM (9*64 -> 32)
    mdconv_gemm<64, 2, 27, false><<<GG, 256, 0, stream>>>(h, nullptr, offw2p, offb2, om2);
    mdconv_gemm<64, 2, 32, true ><<<GG, 256, 0, stream>>>(h, om2,     w2p,    nullptr, gB);

    bn_final_k<<<32, 256, 0, stream>>>(gB, g2, b2, x, gamma, (float*)d_out);
}